// GraphSAGELayer_1864015807059
// MI455X (gfx1250) — compile-verified
//
#include <hip/hip_runtime.h>
#include <hip/hip_bf16.h>
#include <math.h>

typedef __attribute__((ext_vector_type(2))) float        v2f;
typedef __attribute__((ext_vector_type(8))) float        v8f;
typedef __attribute__((ext_vector_type(4))) unsigned int u32x4;
typedef __attribute__((ext_vector_type(4))) int          i32x4;
typedef __attribute__((ext_vector_type(8))) int          i32x8;

#define DIN 128
#define DOUT 128
#define KDIM 256          // 2*DIN
#define LSTRIDE 132       // LDS row stride in floats (128 data + 4 TDM pad)

// ---------------------------------------------------------------- zero scratch
__global__ void sage_zero_kernel(float* __restrict__ ws, long long n) {
    long long i = (long long)blockIdx.x * blockDim.x + threadIdx.x;
    if (i < n) ws[i] = 0.0f;
}

// ---------------------------------------------------------------- edge scatter
// One wave32 per edge: 32 lanes x float4 = 128 features. agg_sum and deg are
// L2-resident (51.2 MB + 0.4 MB << 192 MB L2), so f32 atomics stay on-chip.
__global__ __launch_bounds__(256) void sage_scatter_kernel(
        const float* __restrict__ x, const float* __restrict__ adj,
        const int* __restrict__ row, const int* __restrict__ col,
        float* __restrict__ agg, float* __restrict__ deg, int E) {
    int wid  = (int)((blockIdx.x * blockDim.x + threadIdx.x) >> 5);
    int lane = threadIdx.x & 31;
    if (wid >= E) return;

    int   r  = row[wid];
    int   c  = col[wid];
    float av = adj[wid];

    const float4* xs = (const float4*)(x + (size_t)c * DIN);
    float4 v = xs[lane];

    float* dst = agg + (size_t)r * DIN + lane * 4;
    atomicAdd(dst + 0, av * v.x);
    atomicAdd(dst + 1, av * v.y);
    atomicAdd(dst + 2, av * v.z);
    atomicAdd(dst + 3, av * v.w);
    if (lane == 0) atomicAdd(deg + r, av);
}

// ---------------------------------------------------------------- TDM helper
// 2D tile load Global -> LDS via the Tensor Data Mover (async, TENSORcnt).
// pad: insert 4 DWORDs after every 128 DWORDs -> LDS row stride = 132 floats
// (bank-conflict padding for free, no staging VALU).
// D# field packing per CDNA5 ISA ch.10.8 (groups 0/1; groups 2/3 zero = 2D).
__device__ __forceinline__ void tdm_load_2d(unsigned lds_addr, const void* gptr,
                                            unsigned tensor_d0, unsigned tensor_d1,
                                            unsigned tile_d0,  unsigned tile_d1,
                                            unsigned stride0_elts) {
    unsigned long long ga = (unsigned long long)gptr;
    u32x4 g0;
    g0.x = 1u;                                        // count=1, no gather
    g0.y = lds_addr;                                  // lds_addr[31:0]
    g0.z = (unsigned)(ga & 0xffffffffu);              // global_addr[31:0]
    g0.w = (unsigned)((ga >> 32) & 0x01ffffffu)       // global_addr[56:32]
         | (2u << 30);                                // type = 2 ("image")
    i32x8 g1;
    g1[0] = (int)((2u << 16)                          // data_size = 4B
                | (1u << 20)                          // pad_enable
                | (6u << 22)                          // pad_interval: 128 DW
                | (3u << 25));                        // pad_amount: 4 DW
    g1[1] = (int)((tensor_d0 & 0xffffu) << 16);                        // dim0[15:0]
    g1[2] = (int)(((tensor_d0 >> 16) & 0xffffu)
                | ((tensor_d1 & 0xffffu) << 16));                      // dim0[31:16] | dim1[15:0]
    g1[3] = (int)(((tensor_d1 >> 16) & 0xffffu)
                | ((tile_d0 & 0xffffu) << 16));                        // dim1[31:16] | tile0
    g1[4] = (int)(tile_d1 & 0xffffu);                                  // tile1, tile2=0
    g1[5] = (int)stride0_elts;                                         // dim0_stride[31:0]
    g1[6] = 0;
    g1[7] = 0;
    i32x4 gz = {0, 0, 0, 0};
#if __clang_major__ >= 23
    i32x8 gz8 = {0, 0, 0, 0, 0, 0, 0, 0};
    __builtin_amdgcn_tensor_load_to_lds(g0, g1, gz, gz, gz8, 0);
#else
    __builtin_amdgcn_tensor_load_to_lds(g0, g1, gz, gz, 0);
#endif
}

// ---------------------------------------------------------------- fused GEMM
// Block = 256 threads (8 wave32), handles 16 nodes.
// TDM stages weight [256][128], x-tile [16][128], agg-tile [16][128] into LDS
// (row stride 132). Wave w computes the 16x16 tile for columns [16w,16w+16)
// with V_WMMA_F32_16X16X4_F32: acc1 = x . W_top (+bias), acc2 = agg_sum . W_bot,
// out_row = acc1 + acc2 * (1/deg_row)  [linearity of the mean-aggregation],
// then the block L2-normalizes + ReLUs the 16 rows.
__global__ __launch_bounds__(256) void sage_gemm_kernel(
        const float* __restrict__ x, const float* __restrict__ agg,
        const float* __restrict__ deg, const float* __restrict__ weight,
        const float* __restrict__ bias, float* __restrict__ out, int n) {
    extern __shared__ float smem[];
    float* sX   = smem;                      // [16][LSTRIDE]   x rows
    float* sG   = sX  + 16  * LSTRIDE;       // [16][LSTRIDE]   agg_sum rows
    float* sW   = sG  + 16  * LSTRIDE;       // [256][LSTRIDE]  weight, k-major
    float* sO   = sW  + KDIM * LSTRIDE;      // [16][LSTRIDE]   raw output tile
    float* sDeg = sO  + 16  * LSTRIDE;       // [16]            1/(deg+eps)

    const int t     = threadIdx.x;
    const int node0 = blockIdx.x * 16;

    if ((t >> 5) == 0) {
        // wave 0 drives the three async tensor DMAs
        unsigned ldsX = (unsigned)(size_t)sX;
        unsigned ldsG = (unsigned)(size_t)sG;
        unsigned ldsW = (unsigned)(size_t)sW;
        unsigned rows = (unsigned)(n - node0);          // OOB rows read as zero
        tdm_load_2d(ldsW, weight, DOUT, KDIM, DOUT, KDIM, DOUT);
        tdm_load_2d(ldsX, x   + (size_t)node0 * DIN, DIN, rows, DIN, 16, DIN);
        tdm_load_2d(ldsG, agg + (size_t)node0 * DIN, DIN, rows, DIN, 16, DIN);
        __builtin_amdgcn_s_wait_tensorcnt(0);
    } else if ((t >> 5) == 1) {
        int i = t & 31;
        if (i < 16) {
            int node = node0 + i;
            sDeg[i] = (node < n) ? 1.0f / (deg[node] + 1e-6f) : 0.0f;
        }
    }
    __syncthreads();

    const int wave = t >> 5;     // 0..7 : N-tile index
    const int lane = t & 31;
    const int hl   = lane >> 4;  // half-wave selects K phase (+0 / +2)
    const int ln   = lane & 15;  // M for A operand, N for B/C operands

    // A 16x4 f32 layout: lanes0-15 M=0..15 {K,K+1}; lanes16-31 M=0..15 {K+2,K+3}
    const float* Ax = sX + ln * LSTRIDE + 2 * hl;        // 8B-aligned -> b64
    const float* Ag = sG + ln * LSTRIDE + 2 * hl;
    // B 4x16 f32 layout: lanes0-15 rows K/K+1 at N=0..15; lanes16-31 rows K+2/K+3
    const float* Bb = sW + (2 * hl) * LSTRIDE + wave * 16 + ln;

    v8f acc1, acc2;
    {
        float bv = bias[wave * 16 + ln];   // bias depends only on column N
        #pragma unroll
        for (int i = 0; i < 8; ++i) { acc1[i] = bv; acc2[i] = 0.0f; }
    }

    #pragma unroll 8
    for (int k0 = 0; k0 < DIN; k0 += 4) {
        v2f a1 = *(const v2f*)(Ax + k0);
        v2f b1; b1.x = Bb[k0 * LSTRIDE]; b1.y = Bb[(k0 + 1) * LSTRIDE];
        acc1 = __builtin_amdgcn_wmma_f32_16x16x4_f32(
                   false, a1, false, b1, (short)0, acc1, false, false);
        v2f a2 = *(const v2f*)(Ag + k0);
        v2f b2; b2.x = Bb[(DIN + k0) * LSTRIDE]; b2.y = Bb[(DIN + k0 + 1) * LSTRIDE];
        acc2 = __builtin_amdgcn_wmma_f32_16x16x4_f32(
                   false, a2, false, b2, (short)0, acc2, false, false);
    }

    // C/D layout: VGPR i -> lanes0-15: (M=i, N=lane), lanes16-31: (M=i+8, N)
    #pragma unroll
    for (int i = 0; i < 8; ++i) {
        float dv = sDeg[i + 8 * hl];
        sO[(i + 8 * hl) * LSTRIDE + wave * 16 + ln] = acc1[i] + acc2[i] * dv;
    }
    __syncthreads();

    // ---- row-wise L2 normalize + ReLU; wave handles rows 2w and 2w+1
    #pragma unroll
    for (int rr = 0; rr < 2; ++rr) {
        int mrow = wave * 2 + rr;
        const float4* src = (const float4*)(sO + mrow * LSTRIDE);
        float4 vv = src[lane];                      // 32 lanes x 4 = 128 cols
        float ss = vv.x * vv.x + vv.y * vv.y + vv.z * vv.z + vv.w * vv.w;
        #pragma unroll
        for (int o = 16; o > 0; o >>= 1) ss += __shfl_xor(ss, o, 32);
        float inv = 1.0f / fmaxf(sqrtf(ss), 1e-12f);
        float4 ov;
        ov.x = fmaxf(vv.x * inv, 0.0f);
        ov.y = fmaxf(vv.y * inv, 0.0f);
        ov.z = fmaxf(vv.z * inv, 0.0f);
        ov.w = fmaxf(vv.w * inv, 0.0f);
        if (node0 + mrow < n)
            ((float4*)(out + (size_t)(node0 + mrow) * DOUT))[lane] = ov;
    }
}

// ---------------------------------------------------------------- launch
extern "C" void kernel_launch(void* const* d_in, const int* in_sizes, int n_in,
                              void* d_out, int out_size, void* d_ws, size_t ws_size,
                              hipStream_t stream) {
    const float* x    = (const float*)d_in[0];
    const float* adj  = (const float*)d_in[1];
    const float* wgt  = (const float*)d_in[2];
    const float* bias = (const float*)d_in[3];
    const int*   row  = (const int*)d_in[4];
    const int*   col  = (const int*)d_in[5];
    float*       out  = (float*)d_out;

    const int N = in_sizes[0] / DIN;
    const int E = in_sizes[4];

    float* agg = (float*)d_ws;                 // [N][DIN] raw segment sums
    float* deg = agg + (size_t)N * DIN;        // [N]

    // 1) zero scratch (must happen every call: harness doesn't re-poison)
    long long zn = (long long)N * DIN + N;
    sage_zero_kernel<<<(int)((zn + 255) / 256), 256, 0, stream>>>(agg, zn);

    // 2) edge scatter: one wave32 per edge, 8 edges per 256-thread block
    sage_scatter_kernel<<<(E + 7) / 8, 256, 0, stream>>>(x, adj, row, col,
                                                         agg, deg, E);

    // 3) fused GEMM + normalize + relu: 16 nodes per block, TDM-staged LDS
    size_t smem = (size_t)((16 + 16 + KDIM + 16) * LSTRIDE + 16)
                  * sizeof(float);             // ~157 KB (of 320 KB/WGP)
    sage_gemm_kernel<<<(N + 15) / 16, 256, smem, stream>>>(x, agg, deg, wgt,
                                                           bias, out, N);
}